// deform_LKA3d_68650757259839
// MI455X (gfx1250) — compile-verified
//
#include <hip/hip_runtime.h>

typedef __attribute__((ext_vector_type(2))) float v2f;
typedef __attribute__((ext_vector_type(8))) float v8f;
typedef __attribute__((ext_vector_type(4))) unsigned int u32x4;
typedef __attribute__((ext_vector_type(8))) int i32x8;
typedef __attribute__((ext_vector_type(4))) int i32x4;

#define KS    5
#define PADC  2
#define SDIM  20
#define NC    32
#define K3    125          // 5*5*5
#define MOFF  375          // 3*K3 offset-conv output channels
#define NVOX  8000         // 20^3
#define KDIM  4000         // NC*K3 reduction dim of offset conv
#define K7    343          // 7*7*7

// GEMM tiling for the offset conv
#define BM 128
#define BN 64
#define BK 32

// ---------------------------------------------------------------------------
// Tensor Data Mover: 2D tile load (BM rows x BK floats, row stride KDIM) from
// global into LDS. D# fields per CDNA5 ISA:
//   group0: [1:0]=count=1 | [63:32]=lds_addr | [120:64]=global_addr | type=2
//   group1: [17:16]=data_size(2=4B) | tensor_dim0/1 (remaining extents, for
//           hardware zero-fill OOB) | tile_dim0=BK | tile_dim1=BM |
//           tensor_dim0_stride=KDIM
//   remaining groups: zero (2D tensor, no iterate/gather/multicast)
// This toolchain exposes the 6-arg builtin (clang-23 form):
//   (u32x4 g0, i32x8 g1, i32x4, i32x4, i32x8, i32 cpol)
// Assumption (documented semantics): global_addr is the tile start and
// tensor_dim* give the remaining valid extent from that point; rows past
// tensor_dim1 read as zero, which implements our M-tail padding.
// ---------------------------------------------------------------------------
__device__ __forceinline__ void tdm_load_a_panel(unsigned lds_addr,
                                                 const float* gaddr,
                                                 int rem_k, int rem_m) {
  unsigned long long ga = (unsigned long long)(uintptr_t)gaddr;
  u32x4 g0;
  g0.x = 1u;                                   // count=1, user descriptor
  g0.y = lds_addr;                             // LDS byte address
  g0.z = (unsigned)ga;                         // global_addr[31:0]
  g0.w = (unsigned)(ga >> 32) | (2u << 30);    // global_addr[56:32] | type=2
  unsigned td0 = (unsigned)rem_k;
  unsigned td1 = (unsigned)rem_m;
  i32x8 g1;
  g1[0] = 0x00020000;                          // data_size=4B; no multicast
  g1[1] = (int)((td0 & 0xFFFFu) << 16);        // tensor_dim0 lo16 @ [63:48]
  g1[2] = (int)((td0 >> 16) | ((td1 & 0xFFFFu) << 16)); // td0 hi | td1 lo
  g1[3] = (int)((td1 >> 16) | ((unsigned)BK << 16));    // td1 hi | tile_dim0
  g1[4] = BM;                                  // tile_dim1 | tile_dim2=0
  g1[5] = KDIM;                                // tensor_dim0_stride lo32
  g1[6] = 0;                                   // stride hi | dim1_stride lo
  g1[7] = 0;
  i32x4 gz4 = {0, 0, 0, 0};
  i32x8 gz8 = {0, 0, 0, 0, 0, 0, 0, 0};
  __builtin_amdgcn_tensor_load_to_lds(g0, g1, gz4, gz4, gz8, 0);
}

// ---------------------------------------------------------------------------
// Kernel A: offset conv as implicit GEMM on V_WMMA_F32_16X16X4_F32.
// offs[m,n] = sum_k w_off[m,k] * im2col(x)[k,n] + b_off[m]
//   m in [0,375), k = c*125 + kz*25 + ky*5 + kx, n = z*400 + y*20 + x
//
// Block: 256 thr = 8 waves. Block tile 128(M) x 64(N), BK=32.
// Waves 4(M) x 2(N); each wave owns 32x32 = 2x2 WMMA accumulators.
// A panel: TDM double-buffer (wave 0 issues tensor_load_to_lds for chunk i+1
//          while all waves run chunk i's WMMAs; s_wait_tensorcnt + barrier
//          publishes it). B panel: register-staged im2col gather pipeline.
//
// Fragment layouts per CDNA5 ISA 7.12.2 (f32):
//   A 16x4:  lanes 0-15 hold K=0/1 (vgpr0/1), lanes 16-31 hold K=2/3; M=lane%16
//   B 4x16:  mirrored; N=lane%16
//   C/D:     vgpr v, lane l -> row M = v + 8*(l>>4), col N = l&15
// ---------------------------------------------------------------------------
__global__ __launch_bounds__(256) void offset_conv_wmma(
    const float* __restrict__ x, const float* __restrict__ w_off,
    const float* __restrict__ b_off, float* __restrict__ offs)
{
  __shared__ float As[2][BM * BK];  // [buf][mrow][k]  2 x 16 KB (TDM target)
  __shared__ float Bs[BK * BN];     // [k][ncol]       8 KB

  const int t    = threadIdx.x;
  const int lane = t & 31;
  const int wave = t >> 5;
  const int wm   = (wave & 3) << 5;   // 0,32,64,96
  const int wn   = (wave >> 2) << 5;  // 0,32
  const int n0   = blockIdx.x * BN;
  const int m0   = blockIdx.y * BM;

  // ---- loop-invariant B staging coordinates ----
  // idx = t + i*256 -> col = t&63 (fixed), kk = (t>>6) + i*4
  const int b_col = t & 63;
  const int b_kk0 = t >> 6;
  const int bn    = n0 + b_col;
  const int bz    = bn / 400;
  const int brem  = bn - bz * 400;
  const int by    = brem / 20;
  const int bx    = brem - by * 20;

  float bReg[8];
  auto loadB = [&](int kc) {
#pragma unroll
    for (int i = 0; i < 8; ++i) {
      int k  = kc + b_kk0 + i * 4;
      int c  = k / K3;
      int r  = k - c * K3;
      int kz = r / 25;  r -= kz * 25;
      int ky = r / 5;
      int kx = r - ky * 5;
      int zz = bz - PADC + kz, yy = by - PADC + ky, xx = bx - PADC + kx;
      float v = 0.f;
      if ((unsigned)zz < SDIM && (unsigned)yy < SDIM && (unsigned)xx < SDIM)
        v = x[((c * SDIM + zz) * SDIM + yy) * SDIM + xx];
      bReg[i] = v;
    }
  };

  const float* aBase = w_off + (size_t)m0 * KDIM;
  const int    remM  = MOFF - m0;  // 128 / 128 / 119 : TDM zero-fills the tail

  v8f acc00 = {}, acc01 = {}, acc10 = {}, acc11 = {};

  // ---- prologue: TDM A-panel 0, stage B chunk 0 ----
  if (wave == 0)
    tdm_load_a_panel((unsigned)(uintptr_t)&As[0][0], aBase, KDIM, remM);
  loadB(0);
  if (wave == 0) __builtin_amdgcn_s_wait_tensorcnt(0);
  __syncthreads();   // A panel 0 resident

  for (int kc = 0; kc < KDIM; kc += BK) {
    const int p = (kc >> 5) & 1;

    // commit staged B registers to LDS
#pragma unroll
    for (int i = 0; i < 8; ++i) Bs[t + i * 256] = bReg[i];

    // issue next A panel into the alternate buffer (async on TENSORcnt)
    if (wave == 0 && kc + BK < KDIM)
      tdm_load_a_panel((unsigned)(uintptr_t)&As[p ^ 1][0],
                       aBase + (kc + BK), KDIM - (kc + BK), remM);
    __syncthreads();   // Bs visible; previous readers of As[p^1] are done

    // kick off next B chunk's global gather (overlaps WMMAs)
    if (kc + BK < KDIM) loadB(kc + BK);

    // ---- 8 K-steps x 4 WMMA (2x2 subtiles) ----
    const int kb = (lane >> 4) << 1;   // K pair base per lane half
    const int ml = lane & 15;
    const float* Ap = &As[p][0];
#pragma unroll
    for (int k4 = 0; k4 < BK; k4 += 4) {
      v2f a0, a1, b0, b1;
      a0.x = Ap[(wm + ml)      * BK + k4 + kb];
      a0.y = Ap[(wm + ml)      * BK + k4 + kb + 1];
      a1.x = Ap[(wm + 16 + ml) * BK + k4 + kb];
      a1.y = Ap[(wm + 16 + ml) * BK + k4 + kb + 1];
      b0.x = Bs[(k4 + kb)     * BN + wn + ml];
      b0.y = Bs[(k4 + kb + 1) * BN + wn + ml];
      b1.x = Bs[(k4 + kb)     * BN + wn + 16 + ml];
      b1.y = Bs[(k4 + kb + 1) * BN + wn + 16 + ml];
      acc00 = __builtin_amdgcn_wmma_f32_16x16x4_f32(false, a0, false, b0,
                                                    (short)0, acc00, false, false);
      acc01 = __builtin_amdgcn_wmma_f32_16x16x4_f32(false, a0, false, b1,
                                                    (short)0, acc01, false, false);
      acc10 = __builtin_amdgcn_wmma_f32_16x16x4_f32(false, a1, false, b0,
                                                    (short)0, acc10, false, false);
      acc11 = __builtin_amdgcn_wmma_f32_16x16x4_f32(false, a1, false, b1,
                                                    (short)0, acc11, false, false);
    }

    // next A panel must be resident before anyone proceeds
    if (wave == 0 && kc + BK < KDIM) __builtin_amdgcn_s_wait_tensorcnt(0);
    __syncthreads();
  }

  // ---- store D tiles (+ bias), row-guard for M padding ----
  const int mlo = ((lane >> 4) << 3);  // +0 or +8
  const int nco = lane & 15;
#pragma unroll
  for (int v = 0; v < 8; ++v) {
    int mA = m0 + wm + v + mlo;        // subtile mi=0
    int mB = mA + 16;                  // subtile mi=1
    int nA = n0 + wn + nco;            // subtile ni=0
    int nB = nA + 16;                  // subtile ni=1
    if (mA < MOFF) {
      float bb = b_off[mA];
      offs[mA * NVOX + nA] = acc00[v] + bb;
      offs[mA * NVOX + nB] = acc01[v] + bb;
    }
    if (mB < MOFF) {
      float bb = b_off[mB];
      offs[mB * NVOX + nA] = acc10[v] + bb;
      offs[mB * NVOX + nB] = acc11[v] + bb;
    }
  }
}

// ---------------------------------------------------------------------------
// Kernel B: deformable depthwise sample + weight. One thread per (c, voxel).
// attn1[c,n] = sum_k trilinear(x[c], base(n,k) + off(k,n)) * w_dw[c,k] + b_dw[c]
// ---------------------------------------------------------------------------
__global__ __launch_bounds__(256) void deform_dw(
    const float* __restrict__ x, const float* __restrict__ offs,
    const float* __restrict__ w_dw, const float* __restrict__ b_dw,
    float* __restrict__ attn1)
{
  int tid = blockIdx.x * blockDim.x + threadIdx.x;
  if (tid >= NC * NVOX) return;
  int c = tid / NVOX;
  int n = tid - c * NVOX;
  int z = n / 400; int rem = n - z * 400;
  int y = rem / 20; int xw = rem - y * 20;
  const float* xc = x + c * NVOX;

  float acc = 0.f;
  for (int k = 0; k < K3; ++k) {
    int kz = k / 25, r = k - kz * 25, ky = r / 5, kx = r - ky * 5;
    float pd = (float)(z  - PADC + kz) + offs[(k * 3 + 0) * NVOX + n];
    float ph = (float)(y  - PADC + ky) + offs[(k * 3 + 1) * NVOX + n];
    float pw = (float)(xw - PADC + kx) + offs[(k * 3 + 2) * NVOX + n];
    float d0 = floorf(pd), h0 = floorf(ph), w0 = floorf(pw);
    float fd = pd - d0, fh = ph - h0, fw = pw - w0;
    int di = (int)d0, hi = (int)h0, wi = (int)w0;
    float s = 0.f;
#pragma unroll
    for (int dd = 0; dd < 2; ++dd) {
      int iz = di + dd; float wz = dd ? fd : 1.f - fd;
#pragma unroll
      for (int hh = 0; hh < 2; ++hh) {
        int iy = hi + hh; float wy = hh ? fh : 1.f - fh;
#pragma unroll
        for (int ww = 0; ww < 2; ++ww) {
          int ix = wi + ww; float wx = ww ? fw : 1.f - fw;
          if ((unsigned)iz < SDIM && (unsigned)iy < SDIM && (unsigned)ix < SDIM)
            s += wz * wy * wx * xc[(iz * SDIM + iy) * SDIM + ix];
        }
      }
    }
    acc += s * w_dw[c * K3 + k];
  }
  attn1[tid] = acc + b_dw[c];
}

// ---------------------------------------------------------------------------
// Kernel C: depthwise 7x7x7 conv, dilation 3, pad 9 (same-size output).
// ---------------------------------------------------------------------------
__global__ __launch_bounds__(256) void dw_dilated(
    const float* __restrict__ attn1, const float* __restrict__ w_sp,
    const float* __restrict__ b_sp, float* __restrict__ attn2)
{
  int tid = blockIdx.x * blockDim.x + threadIdx.x;
  if (tid >= NC * NVOX) return;
  int c = tid / NVOX;
  int n = tid - c * NVOX;
  int z = n / 400; int rem = n - z * 400;
  int y = rem / 20; int xw = rem - y * 20;
  const float* ac = attn1 + c * NVOX;
  const float* wc = w_sp + c * K7;

  float acc = 0.f;
  for (int kz = 0; kz < 7; ++kz) {
    int zz = z - 9 + 3 * kz;
    if ((unsigned)zz >= SDIM) continue;
    for (int ky = 0; ky < 7; ++ky) {
      int yy = y - 9 + 3 * ky;
      if ((unsigned)yy >= SDIM) continue;
#pragma unroll
      for (int kx = 0; kx < 7; ++kx) {
        int xx = xw - 9 + 3 * kx;
        if ((unsigned)xx < SDIM)
          acc += ac[(zz * SDIM + yy) * SDIM + xx] * wc[(kz * 7 + ky) * 7 + kx];
      }
    }
  }
  attn2[tid] = acc + b_sp[c];
}

// ---------------------------------------------------------------------------
// Kernel D: pointwise 32x32 GEMM via WMMA + bias, fused final x*attn.
// Block: 128 thr = 4 waves covering full 32(M) x 32(N) per block; K = 32.
// ---------------------------------------------------------------------------
__global__ __launch_bounds__(128) void pointwise_wmma(
    const float* __restrict__ attn2, const float* __restrict__ w_pw,
    const float* __restrict__ b_pw, const float* __restrict__ x,
    float* __restrict__ out)
{
  __shared__ float Asm[32 * 32];  // w_pw [o][c]
  __shared__ float Bsm[32 * 32];  // attn2 [c][ncol]

  const int t    = threadIdx.x;
  const int lane = t & 31;
  const int wave = t >> 5;
  const int wm   = (wave & 1) << 4;
  const int wn   = (wave >> 1) << 4;
  const int n0   = blockIdx.x * 32;

#pragma unroll
  for (int i = 0; i < 8; ++i) {
    int idx = t + i * 128;
    Asm[idx] = w_pw[idx];
    int cc = idx >> 5, col = idx & 31;
    Bsm[idx] = attn2[cc * NVOX + n0 + col];
  }
  __syncthreads();

  v8f acc = {};
  const int kb = (lane >> 4) << 1;
#pragma unroll
  for (int k4 = 0; k4 < 32; k4 += 4) {
    v2f a, b;
    a.x = Asm[(wm + (lane & 15)) * 32 + k4 + kb];
    a.y = Asm[(wm + (lane & 15)) * 32 + k4 + kb + 1];
    b.x = Bsm[(k4 + kb) * 32 + wn + (lane & 15)];
    b.y = Bsm[(k4 + kb + 1) * 32 + wn + (lane & 15)];
    acc = __builtin_amdgcn_wmma_f32_16x16x4_f32(
        false, a, false, b, (short)0, acc, false, false);
  }

#pragma unroll
  for (int v = 0; v < 8; ++v) {
    int o = wm + v + ((lane >> 4) << 3);
    int n = n0 + wn + (lane & 15);
    float val = acc[v] + b_pw[o];
    out[o * NVOX + n] = x[o * NVOX + n] * val;
  }
}

// ---------------------------------------------------------------------------
// Host launcher. Inputs (setup_inputs order):
//  0:x 1:w_off 2:b_off 3:w_dw 4:b_dw 5:w_sp 6:b_sp 7:w_pw 8:b_pw
// Workspace partition (floats): offs[375*8000] | attn1[32*8000] | attn2[32*8000]
//   = 12 MB + 1 MB + 1 MB = 14 MB.
// ---------------------------------------------------------------------------
extern "C" void kernel_launch(void* const* d_in, const int* in_sizes, int n_in,
                              void* d_out, int out_size, void* d_ws, size_t ws_size,
                              hipStream_t stream) {
  (void)in_sizes; (void)n_in; (void)out_size; (void)ws_size;
  const float* x     = (const float*)d_in[0];
  const float* w_off = (const float*)d_in[1];
  const float* b_off = (const float*)d_in[2];
  const float* w_dw  = (const float*)d_in[3];
  const float* b_dw  = (const float*)d_in[4];
  const float* w_sp  = (const float*)d_in[5];
  const float* b_sp  = (const float*)d_in[6];
  const float* w_pw  = (const float*)d_in[7];
  const float* b_pw  = (const float*)d_in[8];
  float* out = (float*)d_out;

  float* ws    = (float*)d_ws;
  float* offs  = ws;                      // 375 * 8000
  float* attn1 = offs + MOFF * NVOX;      // 32 * 8000
  float* attn2 = attn1 + NC * NVOX;       // 32 * 8000

  // A: offset conv (dominant GEMM, fp32 WMMA, TDM A-panels + im2col pipeline)
  offset_conv_wmma<<<dim3(NVOX / BN, (MOFF + BM - 1) / BM), 256, 0, stream>>>(
      x, w_off, b_off, offs);
  // B: deformable depthwise sampling (gather, VALU)
  deform_dw<<<(NC * NVOX + 255) / 256, 256, 0, stream>>>(
      x, offs, w_dw, b_dw, attn1);
  // C: depthwise dilated conv (gather, VALU)
  dw_dilated<<<(NC * NVOX + 255) / 256, 256, 0, stream>>>(
      attn1, w_sp, b_sp, attn2);
  // D: pointwise GEMM (WMMA) + bias + final elementwise x*attn
  pointwise_wmma<<<NVOX / 32, 128, 0, stream>>>(
      attn2, w_pw, b_pw, x, out);
}